// CovidModel_32289564131423
// MI455X (gfx1250) — compile-verified
//
#include <hip/hip_runtime.h>
#include <stdint.h>

// ---------------------------------------------------------------------------
// Covid compartment forecast for MI455X (gfx1250, wave32).
//   A_fc[t,s]  = warm31[s] * exp2( (t+1)*log2(delta[s]) + invT[s]*Lsum[t] )
//   M[t,s]     = sum_{j<32} A_full[t-1-j, s] * (rho[s]*pi[j,s])
// Closed form removes the sequential scan -> fully parallel in (t,s).
// Window evaluated 2 time-steps at once with packed f32 (v_pk_fma_f32).
// ---------------------------------------------------------------------------

#define J_WIN 32
#define NPAIR 16
#define TPB   256
#define CHUNK 128
#define MAXT  1024

typedef unsigned int u32;
typedef u32 u32x4 __attribute__((ext_vector_type(4)));
typedef int i32x8 __attribute__((ext_vector_type(8)));
typedef int i32x4 __attribute__((ext_vector_type(4)));
typedef float f32x2 __attribute__((ext_vector_type(2)));

#if __has_builtin(__builtin_amdgcn_exp2f)
#define FAST_EXP2(x) __builtin_amdgcn_exp2f(x)
#else
#define FAST_EXP2(x) exp2f(x)
#endif

static __device__ __forceinline__ f32x2 fma2(f32x2 a, f32x2 b, f32x2 c) {
#if __has_builtin(__builtin_elementwise_fma)
  return __builtin_elementwise_fma(a, b, c);   // -> v_pk_fma_f32
#else
  f32x2 r; r.x = fmaf(a.x, b.x, c.x); r.y = fmaf(a.y, b.y, c.y); return r;
#endif
}

// ---------------------------------------------------------------------------
// Kernel 1: Lsum[t] = inclusive prefix sum of log2(r_t[k]).  One block,
// Hillis-Steele scan in LDS (10 rounds).  Negligible cost (~1K elements).
// ---------------------------------------------------------------------------
__global__ __launch_bounds__(MAXT) void covid_scan_log2(
    const float* __restrict__ r_t, float* __restrict__ Lsum, int T) {
  __shared__ float buf[MAXT];
  int i = threadIdx.x;
  buf[i] = (i < T) ? log2f(r_t[i]) : 0.0f;
  __syncthreads();
#pragma unroll 1
  for (int off = 1; off < MAXT; off <<= 1) {
    float add = (i >= off) ? buf[i - off] : 0.0f;
    __syncthreads();
    buf[i] += add;
    __syncthreads();
  }
  if (i < T) Lsum[i] = buf[i];
}

// ---------------------------------------------------------------------------
// Kernel 2: one thread per sample, CHUNK=128 days per block row.
// ---------------------------------------------------------------------------
__global__ __launch_bounds__(TPB) void covid_forecast(
    const float* __restrict__ warmup_A,   // [32, S]
    const float* __restrict__ delta,      // [S]
    const float* __restrict__ T_serial,   // [S]
    const float* __restrict__ rho_M,      // [S]
    const float* __restrict__ pi_M,       // [32, S]
    const float* __restrict__ Lsum,       // [T]  (workspace)
    float* __restrict__ M,                // [T, S]
    int T, int S) {
  __shared__ float Ls[MAXT];

  // ---- Stage Lsum into LDS with the Tensor Data Mover (async DMA). ----
#if __has_builtin(__builtin_amdgcn_tensor_load_to_lds)
  if (threadIdx.x == 0) {
    // Low 32 bits of the flat address of a __shared__ object are the LDS
    // byte offset (ISA 10.2: LDS aperture keeps the offset in addr[31:0]).
    u32 lds_addr = (u32)(uintptr_t)(&Ls[0]);
    unsigned long long ga = (unsigned long long)(uintptr_t)Lsum;
    u32x4 g0;
    g0[0] = 1u;                                            // count=1, user D#
    g0[1] = lds_addr;                                      // lds_addr [63:32]
    g0[2] = (u32)ga;                                       // global_addr lo
    g0[3] = ((u32)(ga >> 32) & 0x01FFFFFFu) | (2u << 30);  // ga[56:32], type=2
    i32x8 g1;
    g1[0] = (2 << 16);                          // data_size = 4B; mask/flags 0
    g1[1] = (int)(((u32)T & 0xFFFFu) << 16);    // tensor_dim0 lo16
    g1[2] = (int)(((u32)T >> 16) | (1u << 16)); // tensor_dim0 hi16, tensor_dim1=1
    g1[3] = (int)(((u32)T & 0xFFFFu) << 16);    // tile_dim0 = T
    g1[4] = 1;                                  // tile_dim1 = 1, tile_dim2 = 0
    g1[5] = T;                                  // tensor_dim0_stride lo32
    g1[6] = 0;
    g1[7] = 0;
    i32x4 zz4 = (i32x4)0;                       // D# groups 2/3 unused (1-D)
    i32x8 zz8 = (i32x8)0;
    __builtin_amdgcn_tensor_load_to_lds(g0, g1, zz4, zz4, zz8, 0);
#if __has_builtin(__builtin_amdgcn_s_wait_tensorcnt)
    __builtin_amdgcn_s_wait_tensorcnt(0);
#endif
  }
#else
  for (int i = threadIdx.x; i < T; i += TPB) Ls[i] = Lsum[i];
#endif
  __syncthreads();

  int sid = blockIdx.x * TPB + threadIdx.x;
  bool active = sid < S;
  int s = active ? sid : (S - 1);  // clamp so loads stay in bounds

  float invT   = 1.0f / T_serial[s];
  float ld     = log2f(delta[s]);                      // log2 of daily decay
  float warm31 = warmup_A[(size_t)(J_WIN - 1) * S + s];
  float rho    = rho_M[s];

  float wv[J_WIN];
#pragma unroll
  for (int j = 0; j < J_WIN; ++j) wv[j] = rho * pi_M[(size_t)j * S + s];

  // Packed weight vectors for the two phases of the 2-step window:
  //   M[t]   = sum_q P[q] . W0[q],            W0[q] = (w[31-2q], w[30-2q])
  //   M[t+1] = sum_q P[q] . W1[q] + Pn . Wn,  W1[q] = (wext[32-2q], w[31-2q])
  f32x2 W0[NPAIR], W1[NPAIR];
#pragma unroll
  for (int q = 0; q < NPAIR; ++q) {
    f32x2 t0v = {wv[31 - 2 * q], wv[30 - 2 * q]};
    W0[q] = t0v;
    f32x2 t1v = {(q == 0) ? 0.0f : wv[32 - 2 * q], wv[31 - 2 * q]};
    W1[q] = t1v;
  }
  f32x2 Wn = {wv[0], 0.0f};

  int t0 = blockIdx.y * CHUNK;

  // Ring of 16 day-pairs: P[(i+q)&15] = (A'[t-32+2q], A'[t-31+2q]).
  f32x2 P[NPAIR];
  if (t0 == 0) {
#pragma unroll
    for (int q = 0; q < NPAIR; ++q) {
      f32x2 p = {warmup_A[(size_t)(2 * q) * S + s],
                 warmup_A[(size_t)(2 * q + 1) * S + s]};
      P[q] = p;
    }
  } else {
#pragma unroll
    for (int q = 0; q < NPAIR; ++q) {
      int d = t0 - J_WIN + 2 * q;  // >= 0 since t0 >= CHUNK >= 32
      float e0 = fmaf((float)(d + 1), ld, invT * Ls[d]);
      float e1 = fmaf((float)(d + 2), ld, invT * Ls[d + 1]);
      f32x2 p = {warm31 * FAST_EXP2(e0), warm31 * FAST_EXP2(e1)};
      P[q] = p;
    }
  }

  float* outp = M + (size_t)t0 * S + sid;
#pragma unroll 1
  for (int tb = 0; tb < CHUNK; tb += J_WIN) {
#pragma unroll
    for (int i = 0; i < NPAIR; ++i) {
      int t = t0 + tb + 2 * i;
      // Closed-form A'[t], A'[t+1] (new ring pair).
      float e0 = fmaf((float)(t + 1), ld, invT * Ls[t]);
      float e1 = fmaf((float)(t + 2), ld, invT * Ls[t + 1]);
      f32x2 Pn = {warm31 * FAST_EXP2(e0), warm31 * FAST_EXP2(e1)};

      f32x2 za = {0.0f, 0.0f};
      f32x2 a0 = za, a1 = za, a2 = za, a3 = za;   // M[t]   chains
      f32x2 b0 = za, b1 = za, b2 = za, b3 = za;   // M[t+1] chains
#pragma unroll
      for (int q = 0; q < NPAIR; q += 4) {
        f32x2 p0 = P[(i + q + 0) & (NPAIR - 1)];
        f32x2 p1 = P[(i + q + 1) & (NPAIR - 1)];
        f32x2 p2 = P[(i + q + 2) & (NPAIR - 1)];
        f32x2 p3 = P[(i + q + 3) & (NPAIR - 1)];
        a0 = fma2(p0, W0[q + 0], a0);
        a1 = fma2(p1, W0[q + 1], a1);
        a2 = fma2(p2, W0[q + 2], a2);
        a3 = fma2(p3, W0[q + 3], a3);
        b0 = fma2(p0, W1[q + 0], b0);
        b1 = fma2(p1, W1[q + 1], b1);
        b2 = fma2(p2, W1[q + 2], b2);
        b3 = fma2(p3, W1[q + 3], b3);
      }
      b0 = fma2(Pn, Wn, b0);                       // A'[t]*w[0] term of M[t+1]
      f32x2 sa = (a0 + a1) + (a2 + a3);
      f32x2 sb = (b0 + b1) + (b2 + b3);
      float m0 = sa.x + sa.y;
      float m1 = sb.x + sb.y;

      P[i & (NPAIR - 1)] = Pn;                     // retire oldest pair

      // 205MB output stream > 192MB L2: use non-temporal stores.
      if (active && t < T)     __builtin_nontemporal_store(m0, outp);
      if (active && t + 1 < T) __builtin_nontemporal_store(m1, outp + S);
      outp += 2 * (size_t)S;
    }
  }
}

// ---------------------------------------------------------------------------
extern "C" void kernel_launch(void* const* d_in, const int* in_sizes, int n_in,
                              void* d_out, int out_size, void* d_ws, size_t ws_size,
                              hipStream_t stream) {
  (void)n_in; (void)out_size; (void)ws_size;
  const float* r_t      = (const float*)d_in[0];
  const float* warmup_A = (const float*)d_in[1];
  const float* delta    = (const float*)d_in[2];
  const float* T_serial = (const float*)d_in[3];
  const float* rho_M    = (const float*)d_in[4];
  const float* pi_M     = (const float*)d_in[5];
  float* M = (float*)d_out;
  float* Lsum = (float*)d_ws;   // T floats of scratch

  int T = in_sizes[0];
  int S = in_sizes[2];

  covid_scan_log2<<<1, MAXT, 0, stream>>>(r_t, Lsum, T);

  dim3 grid((S + TPB - 1) / TPB, (T + CHUNK - 1) / CHUNK);
  covid_forecast<<<grid, TPB, 0, stream>>>(warmup_A, delta, T_serial, rho_M,
                                           pi_M, Lsum, M, T, S);
}